// RNNGridLayer_19215683683086
// MI455X (gfx1250) — compile-verified
//
#include <hip/hip_runtime.h>
#include <hip/hip_bf16.h>
#include <math.h>

// ---------------- problem constants (match reference) ----------------
constexpr int Bn  = 4;
constexpr int Cin = 64;
constexpr int Hh  = 128;
constexpr int Ww  = 128;
constexpr int HID = 64;
constexpr int O3  = 3 * HID;     // 192
constexpr int MQ  = 9;           // MOVESQ
constexpr int Ts  = 8;           // TIME_STEPS
constexpr float LOSS_W = 0.01f;
constexpr int HW = Hh * Ww;              // 16384
constexpr int MT_PER_ROW = Ww / 16;      // 8
constexpr int MTILES = Bn * Hh * MT_PER_ROW; // 4096

// d_out layout (floats): hiddens | outs | loss
constexpr size_t OUT_HID  = 0;
constexpr size_t OUT_OUTS = (size_t)Ts * Bn * HID * HW;              // 33,554,432
constexpr size_t OUT_LOSS = OUT_OUTS + (size_t)Ts * Bn * MQ * HW;    // 38,273,024
constexpr size_t LOSS_N   = (size_t)Bn * MQ * HW;                    // 589,824

// d_ws layout (floats): i_all | hy | bce | packed weights
constexpr size_t WS_IALL = 0;
constexpr size_t WS_HY   = (size_t)Bn * O3 * HW;                     // 12,582,912
constexpr size_t WS_BCE  = WS_HY + (size_t)Bn * HID * HW;            // 16,777,216
constexpr size_t WS_WIHP = WS_BCE + 16;                              // 16,777,232
constexpr size_t N_WIHP  = (size_t)12 * 9 * 32 * 16 * 2;             // 110,592
constexpr size_t WS_WHHP = WS_WIHP + N_WIHP;                         // 16,887,824
constexpr size_t N_WHHP  = (size_t)12 * 32 * 16 * 2;                 // 12,288
constexpr size_t WS_WHOP = WS_WHHP + N_WHHP;                         // 16,900,112
constexpr size_t N_WHOP  = (size_t)32 * 16 * 2;                      // 1,024

// ---------------- WMMA f32 16x16x4 ----------------
typedef float v2f __attribute__((ext_vector_type(2)));
typedef float v8f __attribute__((ext_vector_type(8)));

static __device__ __forceinline__ v8f wmma4(v2f a, v2f b, v8f c) {
  // D(16x16,f32) = A(16x4,f32) * B(4x16,f32) + C
  return __builtin_amdgcn_wmma_f32_16x16x4_f32(false, a, false, b, (short)0, c,
                                               false, false);
}

static __device__ __forceinline__ float sigm(float x) {
  return 1.f / (1.f + __expf(-x));
}
static __device__ __forceinline__ int iclamp(int v, int lo, int hi) {
  return v < lo ? lo : (v > hi ? hi : v);
}

// ---------------- init: zero loss region + bce scalar ----------------
__global__ void init_k(float* __restrict__ Loss, float* __restrict__ bce) {
  size_t tid = (size_t)blockIdx.x * blockDim.x + threadIdx.x;
  if (tid < LOSS_N) Loss[tid] = 0.f;
  if (tid == 0) *bce = 0.f;
}

// ---------------- weight packing into WMMA B-fragment order ----------------
// WihP[((ntile*9 + tap)*32 + c2)*16 + n] = {w_ih[ntile*16+n][2*c2][tap], [2*c2+1]}
__global__ void pack_wih(const float* __restrict__ Wih, float* __restrict__ P) {
  int tid = blockIdx.x * blockDim.x + threadIdx.x;     // 12*9*32*16 = 55296
  if (tid >= 12 * 9 * 32 * 16) return;
  int n = tid & 15;
  int c2 = (tid >> 4) & 31;
  int t3 = tid >> 9;
  int tap = t3 % 9;
  int ntile = t3 / 9;
  const float* s = Wih + (size_t)(ntile * 16 + n) * Cin * 9 + (size_t)(2 * c2) * 9 + tap;
  v2f v; v.x = s[0]; v.y = s[9];
  *(v2f*)(P + (size_t)tid * 2) = v;
}

// WhhP[(nt*32 + c2)*16 + n] = {w_hh[nt*16+n][2*c2], [2*c2+1]}
__global__ void pack_whh(const float* __restrict__ Whh, float* __restrict__ P) {
  int tid = blockIdx.x * blockDim.x + threadIdx.x;     // 12*32*16 = 6144
  if (tid >= 12 * 32 * 16) return;
  int n = tid & 15;
  int c2 = (tid >> 4) & 31;
  int nt = tid >> 9;
  const float* s = Whh + (size_t)(nt * 16 + n) * HID + 2 * c2;
  v2f v; v.x = s[0]; v.y = s[1];
  *(v2f*)(P + (size_t)tid * 2) = v;
}

// WhoP[c2*16 + n] = n<9 ? {w_ho[n][2*c2], [2*c2+1]} : 0   (zero-padded N)
__global__ void pack_who(const float* __restrict__ Who, float* __restrict__ P) {
  int tid = blockIdx.x * blockDim.x + threadIdx.x;     // 32*16 = 512
  if (tid >= 32 * 16) return;
  int n = tid & 15;
  int c2 = tid >> 4;
  v2f v = {};
  if (n < MQ) {
    const float* s = Who + (size_t)n * HID + 2 * c2;
    v.x = s[0]; v.y = s[1];
  }
  *(v2f*)(P + (size_t)tid * 2) = v;
}

// ---------------- conv 3x3 (i_all) via WMMA f32 16x16x4 ----------------
// One wave computes a 16-pixel x 16-outchannel tile; K = 9 taps * 64 ch.
// Boundary handling: clamp address, select value (no exec-branching).
__global__ __launch_bounds__(256) void conv_ih_wmma(
    const float* __restrict__ X, const float* __restrict__ WihP,
    const float* __restrict__ bih, float* __restrict__ Iall) {
  int wid  = (int)((blockIdx.x * blockDim.x + threadIdx.x) >> 5);
  int lane = threadIdx.x & 31;
  if (wid >= MTILES * 12) return;           // uniform per wave
  int ntile = wid % 12;
  int mtile = wid / 12;
  int b  = mtile / (Hh * MT_PER_ROW);
  int r  = mtile % (Hh * MT_PER_ROW);
  int y  = r / MT_PER_ROW;
  int x0 = (r % MT_PER_ROW) * 16;
  int m = lane & 15, half = lane >> 4;      // A: row m, k-pair half; B: col n=m

  const v2f* WQ = (const v2f*)WihP;
  v8f acc = {};
  for (int ky = 0; ky < 3; ++ky) {
    int yy = y + ky - 1;
    int yyc = iclamp(yy, 0, Hh - 1);
    bool yok = (unsigned)yy < (unsigned)Hh;
    for (int kx = 0; kx < 3; ++kx) {
      int xx = x0 + m + kx - 1;
      bool ok = yok && ((unsigned)xx < (unsigned)Ww);
      int xxc = iclamp(xx, 0, Ww - 1);
      const float* xp = X + (size_t)b * Cin * HW + (size_t)yyc * Ww + xxc;
      const v2f* wq = WQ + (size_t)(ntile * 9 + ky * 3 + kx) * 32 * 16;
#pragma unroll 4
      for (int c0 = 0; c0 < Cin; c0 += 4) {
        int c = c0 + half * 2;
        float t0 = xp[(size_t)c * HW];
        float t1 = xp[(size_t)(c + 1) * HW];
        v2f a;
        a.x = ok ? t0 : 0.f;
        a.y = ok ? t1 : 0.f;
        v2f bb = wq[(size_t)(c0 / 2 + half) * 16 + m];
        acc = wmma4(a, bb, acc);
      }
    }
  }
  int n = ntile * 16 + m;                   // output channel for this lane
  float bias = bih[n];
  float* op = Iall + (size_t)(b * O3 + n) * HW + (size_t)y * Ww + x0;
#pragma unroll
  for (int v = 0; v < 8; ++v) op[v + half * 8] = acc[v] + bias;
}

// ---------------- per-step GEMM + gates + softmax ----------------
// One wave per 16-pixel tile: hh = hid @ w_hh^T (12 WMMA acc tiles),
// gates -> hy, hy restaged via LDS as A-fragments, logits = hy @ w_ho^T,
// softmax over 9 channels, write hy (ws) and o (outs[t]).
__global__ __launch_bounds__(128) void step_gemm_wmma(
    const float* __restrict__ Hid, const float* __restrict__ Iall,
    const float* __restrict__ WhhP, const float* __restrict__ bhh,
    const float* __restrict__ WhoP, const float* __restrict__ bho,
    float* __restrict__ Hy, float* __restrict__ Outs, int first) {
  __shared__ float sh_hy[4][HID][16];
  __shared__ float sh_lg[4][16][16];
  int wslot = threadIdx.x >> 5;
  int lane  = threadIdx.x & 31;
  int wid   = (int)((blockIdx.x * blockDim.x + threadIdx.x) >> 5);
  int b  = wid / (Hh * MT_PER_ROW);
  int r  = wid % (Hh * MT_PER_ROW);
  int y  = r / MT_PER_ROW;
  int x0 = (r % MT_PER_ROW) * 16;
  int m = lane & 15, half = lane >> 4;
  int pixbase = y * Ww + x0;

  const v2f* WQ = (const v2f*)WhhP;
  const v2f* WO = (const v2f*)WhoP;

  v8f acc[12];
#pragma unroll
  for (int nt = 0; nt < 12; ++nt) acc[nt] = (v8f){};

  if (!first) {
    for (int c0 = 0; c0 < HID; c0 += 4) {
      int c = c0 + half * 2;
      int c2 = c0 / 2 + half;
      const float* hp = Hid + (size_t)(b * HID + c) * HW + pixbase + m;
      v2f a;
      a.x = hp[0];
      a.y = hp[HW];
#pragma unroll
      for (int nt = 0; nt < 12; ++nt) {
        v2f bb = WQ[(size_t)(nt * 32 + c2) * 16 + m];
        acc[nt] = wmma4(a, bb, acc[nt]);
      }
    }
  }

  // gates: h channel tiles nt=0..3 pair with acc[nt]=h_r, acc[nt+4]=h_i, acc[nt+8]=h_n
#pragma unroll
  for (int nt = 0; nt < 4; ++nt) {
    int ch = nt * 16 + m;
    float br = bhh[ch], bi = bhh[HID + ch], bnn = bhh[2 * HID + ch];
    const float* ip = Iall + (size_t)(b * O3 + ch) * HW + pixbase;
    const float* hp = Hid + (size_t)(b * HID + ch) * HW + pixbase;
    float* hw = Hy + (size_t)(b * HID + ch) * HW + pixbase;
#pragma unroll
    for (int v = 0; v < 8; ++v) {
      int mp = v + half * 8;
      float ir = ip[mp];
      float ii = ip[mp + (size_t)HID * HW];
      float in_ = ip[mp + (size_t)2 * HID * HW];
      float rg = sigm(ir + acc[nt][v] + br);
      float ig = sigm(ii + acc[nt + 4][v] + bi);
      float ng = tanhf(in_ + rg * (acc[nt + 8][v] + bnn));
      float hprev = first ? 0.f : hp[mp];
      float hyv = ng + ig * (hprev - ng);
      sh_hy[wslot][ch][mp] = hyv;
      hw[mp] = hyv;
    }
  }

  // logits = hy @ w_ho^T  (N padded 9->16 with zeroed weights in WhoP)
  v8f oacc = {};
  for (int c0 = 0; c0 < HID; c0 += 4) {
    int c = c0 + half * 2;
    int c2 = c0 / 2 + half;
    v2f a, bb;
    a.x = sh_hy[wslot][c][m];
    a.y = sh_hy[wslot][c + 1][m];
    bb = WO[(size_t)c2 * 16 + m];
    oacc = wmma4(a, bb, oacc);
  }
  float lb = (m < MQ) ? bho[m] : 0.f;
#pragma unroll
  for (int v = 0; v < 8; ++v) sh_lg[wslot][v + half * 8][m] = oacc[v] + lb;

  // softmax over 9 channels; lanes 0..15 each own one pixel
  if (lane < 16) {
    int p = lane;
    float mx = -INFINITY;
#pragma unroll
    for (int j = 0; j < MQ; ++j) mx = fmaxf(mx, sh_lg[wslot][p][j]);
    float e[MQ], s = 0.f;
#pragma unroll
    for (int j = 0; j < MQ; ++j) { e[j] = __expf(sh_lg[wslot][p][j] - mx); s += e[j]; }
    float inv = 1.f / s;
    float* op = Outs + (size_t)b * MQ * HW + pixbase + p;
#pragma unroll
    for (int j = 0; j < MQ; ++j) op[(size_t)j * HW] = e[j] * inv;
  }
}

// ---------------- merge: h_next(p) = sum_{dy,dx} hy(p-d) * o_{j(d)}(p-d) ----------------
__global__ __launch_bounds__(256) void merge_k(const float* __restrict__ Hy,
                                               const float* __restrict__ O,
                                               float* __restrict__ Hnext) {
  size_t tid = (size_t)blockIdx.x * blockDim.x + threadIdx.x;
  if (tid >= (size_t)Bn * HID * HW) return;
  int x = tid & (Ww - 1);
  int y = (tid >> 7) & (Hh - 1);
  int c = (tid >> 14) & (HID - 1);
  int b = (int)(tid >> 20);
  float s = 0.f;
#pragma unroll
  for (int j = 0; j < MQ; ++j) {
    int dy = j / 3 - 1, dx = j % 3 - 1;
    int sy = y - dy, sx = x - dx;
    bool ok = (unsigned)sy < (unsigned)Hh && (unsigned)sx < (unsigned)Ww;
    int syc = iclamp(sy, 0, Hh - 1), sxc = iclamp(sx, 0, Ww - 1);
    float hv = Hy[(size_t)(b * HID + c) * HW + syc * Ww + sxc];
    float ov = O[(size_t)(b * MQ + j) * HW + syc * Ww + sxc];
    s += ok ? hv * ov : 0.f;
  }
  Hnext[tid] = s;
}

// ---------------- loss: diff accumulation + BCE reduction ----------------
__global__ __launch_bounds__(256) void loss_k(const float* __restrict__ O,
                                              const float* __restrict__ Mask,
                                              float* __restrict__ Loss,
                                              float* __restrict__ bce_acc) {
  int tid = blockIdx.x * blockDim.x + threadIdx.x;  // Bn*HW threads
  int x = tid & (Ww - 1);
  int y = (tid >> 7) & (Hh - 1);
  int b = tid >> 14;
  int p = y * Ww + x;
  const float* ob = O + (size_t)b * MQ * HW;
  float o4 = ob[4 * HW + p];
#pragma unroll
  for (int j = 0; j < MQ; ++j) {
    int dy = j / 3 - 1, dx = j % 3 - 1;
    int gy = y + dy, gx = x + dx;
    bool ok = (unsigned)gy < (unsigned)Hh && (unsigned)gx < (unsigned)Ww;
    int gyc = iclamp(gy, 0, Hh - 1), gxc = iclamp(gx, 0, Ww - 1);
    float gval = ob[4 * HW + gyc * Ww + gxc];
    float g = ok ? gval : 0.f;
    float oj = ob[(size_t)j * HW + p];
    float d = g - oj;
    Loss[(size_t)(b * MQ + j) * HW + p] += LOSS_W * d * d * oj;
  }
  float mk = Mask[(size_t)b * HW + p];
  float bce = -(mk * __logf(1.f - o4) + (1.f - mk) * __logf(o4)) *
              (1.f / (float)(Bn * HW));
#pragma unroll
  for (int off = 16; off > 0; off >>= 1) bce += __shfl_xor(bce, off, 32);
  if ((threadIdx.x & 31) == 0) atomicAdd(bce_acc, bce);
}

// ---------------- final: broadcast sum of per-step BCE scalars ----------------
__global__ void final_k(float* __restrict__ Loss, const float* __restrict__ bce) {
  size_t tid = (size_t)blockIdx.x * blockDim.x + threadIdx.x;
  if (tid < LOSS_N) Loss[tid] += *bce;
}

extern "C" void kernel_launch(void* const* d_in, const int* in_sizes, int n_in,
                              void* d_out, int out_size, void* d_ws, size_t ws_size,
                              hipStream_t stream) {
  const float* x    = (const float*)d_in[0];
  const float* mask = (const float*)d_in[1];
  const float* wih  = (const float*)d_in[2];
  const float* bih  = (const float*)d_in[3];
  const float* whh  = (const float*)d_in[4];
  const float* bhh  = (const float*)d_in[5];
  const float* who  = (const float*)d_in[6];
  const float* bho  = (const float*)d_in[7];

  float* out  = (float*)d_out;
  float* ws   = (float*)d_ws;
  float* iall = ws + WS_IALL;
  float* hy   = ws + WS_HY;
  float* bce  = ws + WS_BCE;
  float* wihp = ws + WS_WIHP;
  float* whhp = ws + WS_WHHP;
  float* whop = ws + WS_WHOP;
  float* loss = out + OUT_LOSS;

  // 0) pack weights into WMMA B-fragment order (b64-coalesced loads)
  pack_wih<<<(12 * 9 * 32 * 16 + 255) / 256, 256, 0, stream>>>(wih, wihp);
  pack_whh<<<(12 * 32 * 16 + 255) / 256, 256, 0, stream>>>(whh, whhp);
  pack_who<<<(32 * 16 + 255) / 256, 256, 0, stream>>>(who, whop);

  // 1) zero loss accumulator + bce scalar
  init_k<<<(unsigned)((LOSS_N + 255) / 256), 256, 0, stream>>>(loss, bce);

  // 2) conv i_all  (4096 M-tiles x 12 N-tiles, 1 wave each)
  conv_ih_wmma<<<(MTILES * 12 * 32) / 256, 256, 0, stream>>>(x, wihp, bih, iall);

  // 3) recurrent steps
  for (int t = 0; t < Ts; ++t) {
    const float* hid = (t == 0) ? out /*unused, guarded by first*/
                                : out + OUT_HID + (size_t)(t - 1) * Bn * HID * HW;
    float* outs_t = out + OUT_OUTS + (size_t)t * Bn * MQ * HW;
    float* hnext  = out + OUT_HID + (size_t)t * Bn * HID * HW;

    step_gemm_wmma<<<(MTILES * 32) / 128, 128, 0, stream>>>(
        hid, iall, whhp, bhh, whop, bho, hy, outs_t, (t == 0) ? 1 : 0);

    merge_k<<<(Bn * HID * HW) / 256, 256, 0, stream>>>(hy, outs_t, hnext);

    loss_k<<<(Bn * HW) / 256, 256, 0, stream>>>(outs_t, mask, loss, bce);
  }

  // 4) broadcast accumulated BCE scalars into the loss image
  final_k<<<(unsigned)((LOSS_N + 255) / 256), 256, 0, stream>>>(loss, bce);
}